// HeteroGNN_38706245272172
// MI455X (gfx1250) — compile-verified
//
#include <hip/hip_runtime.h>
#include <hip/hip_bf16.h>
#include <stdint.h>

#define N_NODES 50000
#define DIM     128       // feature / hidden dim (GEMM K)
#define HID     128       // hidden channels (GEMM N)
#define OUT_DIM 64        // head out channels
#define LSLOPE  0.01f

typedef float v2f __attribute__((ext_vector_type(2)));
typedef float v8f __attribute__((ext_vector_type(8)));

// ---------------------------------------------------------------------------
// CSR build, pass 1: per-dst degree counts (int atomics, 1 per edge).
// ---------------------------------------------------------------------------
__global__ __launch_bounds__(256)
void edge_count(const long long* __restrict__ ei, int* __restrict__ cnt, int nedges)
{
    const int stride = (int)(gridDim.x * blockDim.x);
    for (int e = (int)(blockIdx.x * blockDim.x + threadIdx.x); e < nedges; e += stride)
        atomicAdd(&cnt[(int)ei[(long long)nedges + e]], 1);
}

// ---------------------------------------------------------------------------
// CSR build, pass 2: single-workgroup exclusive scan over N counts (LDS scan).
// Writes both off[] (persistent) and cur[] (fill cursors).
// ---------------------------------------------------------------------------
__global__ __launch_bounds__(1024)
void exscan_kernel(const int* __restrict__ cnt, int* __restrict__ off,
                   int* __restrict__ cur, int n)
{
    __shared__ int sums[1024];
    __shared__ int carry;
    if (threadIdx.x == 0) carry = 0;
    __syncthreads();
    for (int base = 0; base < n; base += 1024) {
        const int i = base + (int)threadIdx.x;
        const int v = (i < n) ? cnt[i] : 0;
        sums[threadIdx.x] = v;
        __syncthreads();
        for (int s = 1; s < 1024; s <<= 1) {
            int t = 0;
            if ((int)threadIdx.x >= s) t = sums[threadIdx.x - s];
            __syncthreads();
            if ((int)threadIdx.x >= s) sums[threadIdx.x] += t;
            __syncthreads();
        }
        const int excl = carry + sums[threadIdx.x] - v;
        if (i < n) { off[i] = excl; cur[i] = excl; }
        __syncthreads();
        if (threadIdx.x == 1023) carry += sums[1023];
        __syncthreads();
    }
    if (threadIdx.x == 0) off[n] = carry;
}

// ---------------------------------------------------------------------------
// CSR build, pass 3: scatter src ids into per-dst segments.
// ---------------------------------------------------------------------------
__global__ __launch_bounds__(256)
void edge_fill(const long long* __restrict__ ei, int* __restrict__ cur,
               int* __restrict__ srcidx, int nedges)
{
    const int stride = (int)(gridDim.x * blockDim.x);
    for (int e = (int)(blockIdx.x * blockDim.x + threadIdx.x); e < nedges; e += stride) {
        const int d = (int)ei[(long long)nedges + e];
        const int s = (int)ei[e];
        const int pos = atomicAdd(&cur[d], 1);
        srcidx[pos] = s;
    }
}

// ---------------------------------------------------------------------------
// Gather-mean: one wave32 per dst node. Each lane owns 4 consecutive floats
// of the 128-wide row (float4, 512B/row coalesced; tables are L2-resident).
// Neighbor ids loaded 32-at-a-time and broadcast via __shfl (DS permute).
// Accumulation in registers -> zero float atomics. mean=0 for isolated nodes.
// ---------------------------------------------------------------------------
__global__ __launch_bounds__(256)
void gather_mean(const float* __restrict__ xsrc, const int* __restrict__ off,
                 const int* __restrict__ srcidx, float* __restrict__ mean, int nnodes)
{
    const int lane = threadIdx.x & 31;
    const int nwav = (int)((gridDim.x * blockDim.x) >> 5);
    for (int d = (int)((blockIdx.x * blockDim.x + threadIdx.x) >> 5); d < nnodes; d += nwav) {
        const int s0 = off[d], s1 = off[d + 1];
        float ax = 0.f, ay = 0.f, az = 0.f, aw = 0.f;
        for (int base = s0; base < s1; base += 32) {
            const int nb = min(32, s1 - base);
            const int myidx = (lane < nb) ? srcidx[base + lane] : 0;
            for (int j = 0; j < nb; ++j) {
                const int s = __shfl(myidx, j, 32);
                const float4 v = ((const float4*)(xsrc + (size_t)s * DIM))[lane];
                ax += v.x; ay += v.y; az += v.z; aw += v.w;
            }
        }
        const float inv = (s1 > s0) ? 1.0f / (float)(s1 - s0) : 0.0f;
        float4 r;
        r.x = ax * inv; r.y = ay * inv; r.z = az * inv; r.w = aw * inv;
        ((float4*)(mean + (size_t)d * DIM))[lane] = r;
    }
}

// ---------------------------------------------------------------------------
// SAGE update:  out = act( mean @ Wl + root @ Wr + bias )
// One wave32 per 16x16 output tile, exact-FP32 WMMA (16x16x4), K=128 x2.
// A fragment: lane(0..15)=row M, lane[4]=hi selects K pair {2*hi, 2*hi+1}.
// C fragment: vgpr r -> M = r + 8*hi, N = lane&15.
// ---------------------------------------------------------------------------
template<bool ACT>
__global__ __launch_bounds__(256)
void sage_update(const float* __restrict__ Amean, const float* __restrict__ Aroot,
                 const float* __restrict__ Wl, const float* __restrict__ Wr,
                 const float* __restrict__ bias, float* __restrict__ out)
{
    const int lane = threadIdx.x & 31;
    const int wv   = threadIdx.x >> 5;     // 8 waves -> 8 N-tiles of HID=128
    const int rc   = lane & 15;
    const int hi   = lane >> 4;
    const int m0   = blockIdx.x * 16;
    const int n0   = wv * 16;

    v8f c = {0.f, 0.f, 0.f, 0.f, 0.f, 0.f, 0.f, 0.f};

    const float* arow = Amean + (size_t)(m0 + rc) * DIM;
    const float* rrow = Aroot + (size_t)(m0 + rc) * DIM;

    #pragma unroll
    for (int k0 = 0; k0 < DIM; k0 += 4) {
        const int ka = k0 + 2 * hi;
        v2f a, b;
        a.x = arow[ka];
        a.y = arow[ka + 1];
        b.x = Wl[(size_t)ka       * HID + n0 + rc];
        b.y = Wl[(size_t)(ka + 1) * HID + n0 + rc];
        c = __builtin_amdgcn_wmma_f32_16x16x4_f32(false, a, false, b,
                                                  (short)0, c, false, false);
    }
    #pragma unroll
    for (int k0 = 0; k0 < DIM; k0 += 4) {
        const int ka = k0 + 2 * hi;
        v2f a, b;
        a.x = rrow[ka];
        a.y = rrow[ka + 1];
        b.x = Wr[(size_t)ka       * HID + n0 + rc];
        b.y = Wr[(size_t)(ka + 1) * HID + n0 + rc];
        c = __builtin_amdgcn_wmma_f32_16x16x4_f32(false, a, false, b,
                                                  (short)0, c, false, false);
    }

    const float bv = bias[n0 + rc];
    #pragma unroll
    for (int r = 0; r < 8; ++r) {
        const int m = m0 + r + 8 * hi;
        float v = c[r] + bv;
        if (ACT) v = (v > 0.0f) ? v : v * LSLOPE;
        out[(size_t)m * HID + n0 + rc] = v;
    }
}

// ---------------------------------------------------------------------------
// Output head: out = X @ W_out + b_out   ([N,128] @ [128,64])
// ---------------------------------------------------------------------------
__global__ __launch_bounds__(128)
void head_gemm(const float* __restrict__ X, const float* __restrict__ W,
               const float* __restrict__ bias, float* __restrict__ out)
{
    const int lane = threadIdx.x & 31;
    const int wv   = threadIdx.x >> 5;     // 0..3
    const int rc   = lane & 15;
    const int hi   = lane >> 4;
    const int m0   = blockIdx.x * 16;
    const int n0   = wv * 16;

    v8f c = {0.f, 0.f, 0.f, 0.f, 0.f, 0.f, 0.f, 0.f};
    const float* xrow = X + (size_t)(m0 + rc) * HID;

    #pragma unroll
    for (int k0 = 0; k0 < HID; k0 += 4) {
        const int ka = k0 + 2 * hi;
        v2f a, b;
        a.x = xrow[ka];
        a.y = xrow[ka + 1];
        b.x = W[(size_t)ka       * OUT_DIM + n0 + rc];
        b.y = W[(size_t)(ka + 1) * OUT_DIM + n0 + rc];
        c = __builtin_amdgcn_wmma_f32_16x16x4_f32(false, a, false, b,
                                                  (short)0, c, false, false);
    }

    const float bv = bias[n0 + rc];
    #pragma unroll
    for (int r = 0; r < 8; ++r) {
        out[(size_t)(m0 + r + 8 * hi) * OUT_DIM + n0 + rc] = c[r] + bv;
    }
}

// ---------------------------------------------------------------------------
extern "C" void kernel_launch(void* const* d_in, const int* in_sizes, int n_in,
                              void* d_out, int out_size, void* d_ws, size_t ws_size,
                              hipStream_t stream)
{
    (void)n_in; (void)out_size; (void)ws_size;

    const float* x_user   = (const float*)d_in[0];
    const float* x_item   = (const float*)d_in[1];
    const float* W0_u2i_l = (const float*)d_in[2];
    const float* W0_u2i_r = (const float*)d_in[3];
    const float* b0_u2i   = (const float*)d_in[4];
    const float* W0_i2u_l = (const float*)d_in[5];
    const float* W0_i2u_r = (const float*)d_in[6];
    const float* b0_i2u   = (const float*)d_in[7];
    // d_in[8..10] = W1_u2i_* / b1_u2i : unused (reference only needs xu2)
    const float* W1_i2u_l = (const float*)d_in[11];
    const float* W1_i2u_r = (const float*)d_in[12];
    const float* b1_i2u   = (const float*)d_in[13];
    const float* W_outp   = (const float*)d_in[14];
    const float* b_outp   = (const float*)d_in[15];
    const long long* ei_u2i = (const long long*)d_in[16];
    const long long* ei_i2u = (const long long*)d_in[17];
    const int E = in_sizes[16] / 2;

    // ---- workspace layout ----
    float* ws = (float*)d_ws;
    const size_t feat = (size_t)N_NODES * DIM;
    float* meanU = ws;                     // [N,128] neighbor means -> users
    float* meanI = meanU + feat;           // [N,128] neighbor means -> items
    float* xu    = meanI + feat;           // layer-0 user hidden
    float* xi    = xu + feat;              // layer-0 item hidden
    int* ib   = (int*)(xi + feat);
    int* cntU = ib;                        // [N]   i2u degree counts
    int* cntI = cntU + N_NODES;            // [N]   u2i degree counts
    int* offU = cntI + N_NODES;            // [N+1]
    int* curU = offU + N_NODES + 1;        // [N]
    int* offI = curU + N_NODES;            // [N+1]
    int* curI = offI + N_NODES + 1;        // [N]
    int* srcU = curI + N_NODES;            // [E]   i2u CSR src ids
    int* srcI = srcU + E;                  // [E]   u2i CSR src ids
    float* xu2 = meanI;                    // reuse after layer 0

    // zero degree counters (capture-safe memset node)
    hipMemsetAsync(cntU, 0, 2 * (size_t)N_NODES * sizeof(int), stream);

    const int EB = 6144;   // edge-parallel blocks
    const int GB = (N_NODES * 32 + 255) / 256;   // one wave per node

    // ---- build dst-indexed CSR for both edge types ----
    edge_count<<<EB, 256, 0, stream>>>(ei_i2u, cntU, E);
    edge_count<<<EB, 256, 0, stream>>>(ei_u2i, cntI, E);
    exscan_kernel<<<1, 1024, 0, stream>>>(cntU, offU, curU, N_NODES);
    exscan_kernel<<<1, 1024, 0, stream>>>(cntI, offI, curI, N_NODES);
    edge_fill<<<EB, 256, 0, stream>>>(ei_i2u, curU, srcU, E);
    edge_fill<<<EB, 256, 0, stream>>>(ei_u2i, curI, srcI, E);

    // ---- layer 0 ----
    gather_mean<<<GB, 256, 0, stream>>>(x_item, offU, srcU, meanU, N_NODES);
    gather_mean<<<GB, 256, 0, stream>>>(x_user, offI, srcI, meanI, N_NODES);

    const int MT = N_NODES / 16;           // 3125, exact
    sage_update<true><<<MT, 256, 0, stream>>>(meanU, x_user,
                                              W0_i2u_l, W0_i2u_r, b0_i2u, xu);
    sage_update<true><<<MT, 256, 0, stream>>>(meanI, x_item,
                                              W0_u2i_l, W0_u2i_r, b0_u2i, xi);

    // ---- layer 1 (users only) ----
    gather_mean<<<GB, 256, 0, stream>>>(xi, offU, srcU, meanU, N_NODES);
    sage_update<true><<<MT, 256, 0, stream>>>(meanU, xu,
                                              W1_i2u_l, W1_i2u_r, b1_i2u, xu2);

    // ---- output head ----
    head_gemm<<<MT, 128, 0, stream>>>(xu2, W_outp, b_outp, (float*)d_out);
}